// QueryAndGroup_grouped_density_Ellipsoid_78374563217944
// MI455X (gfx1250) — compile-verified
//
#include <hip/hip_runtime.h>

// Problem constants (fixed by the reference file)
#define BB 4
#define NN 8192
#define PP 2048
#define CC 64
#define SS 32

typedef __attribute__((ext_vector_type(2))) float v2f;
typedef __attribute__((ext_vector_type(8))) float v8f;

__device__ __forceinline__ unsigned ballot32(bool p) {
#if __has_builtin(__builtin_amdgcn_ballot_w32)
  return __builtin_amdgcn_ballot_w32(p);
#else
  return (unsigned)__ballot(p);
#endif
}

// Select one of 8 wave-uniform values by per-lane index v (0..7): 7 v_cndmask ops.
__device__ __forceinline__ unsigned sel8(unsigned b0, unsigned b1, unsigned b2, unsigned b3,
                                         unsigned b4, unsigned b5, unsigned b6, unsigned b7,
                                         unsigned v) {
  unsigned a0 = (v & 1) ? b1 : b0;
  unsigned a1 = (v & 1) ? b3 : b2;
  unsigned a2 = (v & 1) ? b5 : b4;
  unsigned a3 = (v & 1) ? b7 : b6;
  unsigned c0 = (v & 2) ? a1 : a0;
  unsigned c1 = (v & 2) ? a3 : a2;
  return (v & 4) ? c1 : c0;
}

struct P3 { float x, y, z; };

// ---------------------------------------------------------------------------
// Kernel A: ellipsoid query via V_WMMA_F32_16X16X4_F32 with C = inline 0.
// One wave owns 32 query points (two 16-row A tiles, loop-invariant), and
// streams the point cloud 32 points per iteration (two B subtiles, 4 WMMAs).
//   D[m,n] = A.B = -2 a'.b' + |a'|^2     (a' = a*[5,2.5,5])
//   hit  <=>  q < 1  <=>  D[m,n] < 1 - |b'_n|^2   (per-column threshold;
//   in the D layout each lane holds exactly one column, whose norm it
//   already computed -> no C-matrix broadcast needed at all).
// ballot(D[v] < th) yields two wave-uniform 16-bit row masks per VGPR; each
// lane assembles its own query's 32-bit mask and consumes it with a ctz loop
// (ascending index order => reference's "first 32 hits" semantics).
// Point loads are software-pipelined one tile ahead to hide VMEM latency
// (only ~2 waves/SIMD exist, so latency cannot be hidden by TLP).
// ---------------------------------------------------------------------------
__global__ __launch_bounds__(64) void eq_idx_kernel(const float* __restrict__ xyz,
                                                    const float* __restrict__ new_xyz,
                                                    int* __restrict__ idx) {
  const int lane = threadIdx.x & 31;
  const int wave = blockIdx.x * (blockDim.x >> 5) + (threadIdx.x >> 5);
  const int qbase = wave * 32;              // global query index base (b*PP + p)
  const int b = qbase / PP;
  const bool hi = lane >= 16;

  const float SX = 5.0f;   // 1/r
  const float SY = 2.5f;   // 1/(2r)
  const float SZ = 5.0f;

  // A tiles: lane l supplies query (qbase + 16*t + (l&15)); K = 2*(l/16)+v
  //   lanes<16: {x', y'}   lanes>=16: {z', |a'|^2}
  const float* q0p = new_xyz + (size_t)(qbase + (lane & 15)) * 3;
  const float* q1p = q0p + 16 * 3;
  float ax0 = q0p[0] * SX, ay0 = q0p[1] * SY, az0 = q0p[2] * SZ;
  float ax1 = q1p[0] * SX, ay1 = q1p[1] * SY, az1 = q1p[2] * SZ;
  float na0 = ax0 * ax0 + ay0 * ay0 + az0 * az0;
  float na1 = ax1 * ax1 + ay1 * ay1 + az1 * az1;
  v2f A0, A1;
  A0.x = hi ? az0 : ax0;  A0.y = hi ? na0 : ay0;
  A1.x = hi ? az1 : ax1;  A1.y = hi ? na1 : ay1;

  const v8f Zc = {};                        // C = 0 -> inline-constant SRC2

  const float* xb = xyz + (size_t)b * NN * 3;
  int* myidx = idx + (size_t)(qbase + lane) * SS;

  // Software pipeline: preload tile 0 (points [0,32) for this lane's column).
  const float* c0 = xb + (size_t)(lane & 15) * 3;
  P3 nxt0 = { c0[0], c0[1], c0[2] };
  P3 nxt1 = { c0[48], c0[49], c0[50] };          // +16 points = +48 floats

  int cnt = 0, first = 0;
  for (int nb = 0; nb < NN; nb += 32) {
    P3 cur0 = nxt0, cur1 = nxt1;
    if (nb + 32 < NN) {                          // issue next tile's loads early
      const float* np = xb + (size_t)(nb + 32 + (lane & 15)) * 3;
      nxt0.x = np[0];  nxt0.y = np[1];  nxt0.z = np[2];
      nxt1.x = np[48]; nxt1.y = np[49]; nxt1.z = np[50];
      __builtin_prefetch(np + 96, 0, 0);         // L2 prefetch two tiles ahead
    }

    // ---- subtile 0: points nb .. nb+15 ----
    float bx = cur0.x * SX, by = cur0.y * SY, bz = cur0.z * SZ;
    float th0 = 1.0f - (bx * bx + by * by + bz * bz);   // per-column threshold
    v2f Bm;
    Bm.x = hi ? (-2.0f * bz) : (-2.0f * bx);
    Bm.y = hi ? 1.0f : (-2.0f * by);
    v8f D0 = __builtin_amdgcn_wmma_f32_16x16x4_f32(false, A0, false, Bm, (short)0, Zc, false, false);
    v8f D1 = __builtin_amdgcn_wmma_f32_16x16x4_f32(false, A1, false, Bm, (short)0, Zc, false, false);

    // ---- subtile 1: points nb+16 .. nb+31 ----
    float bx2 = cur1.x * SX, by2 = cur1.y * SY, bz2 = cur1.z * SZ;
    float th1 = 1.0f - (bx2 * bx2 + by2 * by2 + bz2 * bz2);
    v2f Bm2;
    Bm2.x = hi ? (-2.0f * bz2) : (-2.0f * bx2);
    Bm2.y = hi ? 1.0f : (-2.0f * by2);
    v8f E0 = __builtin_amdgcn_wmma_f32_16x16x4_f32(false, A0, false, Bm2, (short)0, Zc, false, false);
    v8f E1 = __builtin_amdgcn_wmma_f32_16x16x4_f32(false, A1, false, Bm2, (short)0, Zc, false, false);

    // Wave-uniform hit masks: bal[v] bits[15:0] = row M=v, bits[31:16] = M=v+8.
    unsigned p00 = ballot32(D0[0] < th0), p01 = ballot32(D0[1] < th0);
    unsigned p02 = ballot32(D0[2] < th0), p03 = ballot32(D0[3] < th0);
    unsigned p04 = ballot32(D0[4] < th0), p05 = ballot32(D0[5] < th0);
    unsigned p06 = ballot32(D0[6] < th0), p07 = ballot32(D0[7] < th0);
    unsigned p10 = ballot32(D1[0] < th0), p11 = ballot32(D1[1] < th0);
    unsigned p12 = ballot32(D1[2] < th0), p13 = ballot32(D1[3] < th0);
    unsigned p14 = ballot32(D1[4] < th0), p15 = ballot32(D1[5] < th0);
    unsigned p16 = ballot32(D1[6] < th0), p17 = ballot32(D1[7] < th0);
    unsigned q00 = ballot32(E0[0] < th1), q01 = ballot32(E0[1] < th1);
    unsigned q02 = ballot32(E0[2] < th1), q03 = ballot32(E0[3] < th1);
    unsigned q04 = ballot32(E0[4] < th1), q05 = ballot32(E0[5] < th1);
    unsigned q06 = ballot32(E0[6] < th1), q07 = ballot32(E0[7] < th1);
    unsigned q10 = ballot32(E1[0] < th1), q11 = ballot32(E1[1] < th1);
    unsigned q12 = ballot32(E1[2] < th1), q13 = ballot32(E1[3] < th1);
    unsigned q14 = ballot32(E1[4] < th1), q15 = ballot32(E1[5] < th1);
    unsigned q16 = ballot32(E1[6] < th1), q17 = ballot32(E1[7] < th1);

    // Lane i owns query qbase+i: tile t=i>>4, VGPR v=i&7, half shift h=(i&8)*2.
    unsigned v = (unsigned)(lane & 7);
    unsigned h = (unsigned)((lane & 8) << 1);
    unsigned sa = sel8(p00, p01, p02, p03, p04, p05, p06, p07, v);
    unsigned sb = sel8(p10, p11, p12, p13, p14, p15, p16, p17, v);
    unsigned ta = sel8(q00, q01, q02, q03, q04, q05, q06, q07, v);
    unsigned tb = sel8(q10, q11, q12, q13, q14, q15, q16, q17, v);
    unsigned mlo = ((((lane & 16) ? sb : sa) >> h) & 0xffffu);
    unsigned mhi = ((((lane & 16) ? tb : ta) >> h) & 0xffffu);
    unsigned mask = mlo | (mhi << 16);           // bit k -> point nb + k

    while (mask) {
      if (cnt >= SS) break;
      int bit = (int)__builtin_ctz(mask);
      mask &= mask - 1;
      int n = nb + bit;
      if (cnt == 0) first = n;
      myidx[cnt] = n;
      ++cnt;
    }
    if (ballot32(cnt >= SS) == 0xffffffffu) break;   // uniform early exit
  }
  for (int s = cnt; s < SS; ++s) myidx[s] = first;   // pad with first hit (0 if none)
}

// ---------------------------------------------------------------------------
// Kernel B: gather + density.  One wave per (b,p); lane = sample s.
// Channel-major writes are lane-contiguous (128B per wave store).  Output is
// write-once -> non-temporal stores keep L2 free for the gathered feature
// table (8 MB) and xyz, which are re-read ~8x each.
// ---------------------------------------------------------------------------
__global__ __launch_bounds__(256) void gather_kernel(const float* __restrict__ xyz,
                                                     const float* __restrict__ new_xyz,
                                                     const float* __restrict__ feat,
                                                     const int* __restrict__ idx,
                                                     float* __restrict__ out) {
  const int lane = threadIdx.x & 31;
  const int gw = blockIdx.x * (blockDim.x >> 5) + (threadIdx.x >> 5);  // 0..BB*PP-1
  const int b = gw / PP;
  const int p = gw - b * PP;

  const int n = idx[(size_t)gw * SS + lane];

  const float* xp = xyz + ((size_t)b * NN + n) * 3;
  float gx = xp[0], gy = xp[1], gz = xp[2];
  const float* qp = new_xyz + (size_t)gw * 3;
  float dx = gx - qp[0], dy = gy - qp[1], dz = gz - qp[2];
  float d2 = dx * dx + dy * dy + dz * dz;
  // 1/density = 2.5*BW * exp(d^2 / (2*BW^2)) with BW=0.1
  float invdens = 0.25f * expf(50.0f * d2);

  const size_t chs = (size_t)PP * SS;                       // channel stride
  float* ob = out + ((size_t)b * 68 * PP + p) * SS + lane;
  __builtin_nontemporal_store(gx, ob + 0 * chs);
  __builtin_nontemporal_store(gy, ob + 1 * chs);
  __builtin_nontemporal_store(gz, ob + 2 * chs);
  __builtin_nontemporal_store(invdens, ob + 3 * chs);

  const float* fb = feat + (size_t)b * CC * NN + n;
  float* of = ob + 4 * chs;
#pragma unroll 8
  for (int c = 0; c < CC; ++c)
    __builtin_nontemporal_store(fb[(size_t)c * NN], of + (size_t)c * chs);
}

extern "C" void kernel_launch(void* const* d_in, const int* in_sizes, int n_in,
                              void* d_out, int out_size, void* d_ws, size_t ws_size,
                              hipStream_t stream) {
  const float* xyz     = (const float*)d_in[0];   // (4, 8192, 3)
  const float* new_xyz = (const float*)d_in[1];   // (4, 2048, 3)
  const float* feat    = (const float*)d_in[2];   // (4, 64, 8192)
  float* out = (float*)d_out;                     // (4, 68, 2048, 32)
  int* idx = (int*)d_ws;                          // (4*2048, 32) = 1 MB scratch

  // Query: 256 waves total (32 queries each), 2 waves/block -> 128 blocks.
  eq_idx_kernel<<<dim3((BB * PP) / (2 * 32)), dim3(64), 0, stream>>>(xyz, new_xyz, idx);
  // Gather: one wave per (b,p) -> 8192 waves, 8 waves/block -> 1024 blocks.
  gather_kernel<<<dim3((BB * PP) / 8), dim3(256), 0, stream>>>(xyz, new_xyz, feat, idx, out);
}